// GATv2AttentionPool_10806137717386
// MI455X (gfx1250) — compile-verified
//
#include <hip/hip_runtime.h>

#define HH 8
#define CC 8
#define HC 64

typedef __attribute__((ext_vector_type(2))) float v2f;
typedef __attribute__((ext_vector_type(8))) float v8f;

// Order-preserving float -> uint map (total order incl. negatives) so we can
// do segment-max with plain global_atomic_max_u32 (no CAS loop).
__device__ __forceinline__ unsigned fmap(float f) {
    int i = __float_as_int(f);
    return (unsigned)(i ^ ((i >> 31) | (int)0x80000000));
}
__device__ __forceinline__ float funmap(unsigned u) {
    int i = (u & 0x80000000u) ? (int)(u ^ 0x80000000u) : (int)(~u);
    return __int_as_float(i);
}

// ---------------- K1: init accumulators ----------------
__global__ void k_init(float* __restrict__ out, unsigned* __restrict__ segmax,
                       float* __restrict__ denom, int nOut, int nNH) {
    int i = blockIdx.x * blockDim.x + threadIdx.x;
    if (i < nOut) out[i] = 0.0f;
    if (i < nNH) { segmax[i] = 0u; denom[i] = 0.0f; }
}

// ---------------- K2: logits via V_WMMA_F32_16X16X4_F32 + scatter max ----
// Per 16-edge tile: D[m, h] = sum_kg feats[e0+m][kg] * B[kg][h],
// B[kg][h] = (h == kg/8) ? ak[kg%8][h] : 0  -> exactly the per-head dot.
// B depends only on (kk, lane): build all 16 fragments once per wave, then
// grid-stride over tiles so the inner loop is pure load->pk_add->wmma.
__global__ void k_logits(const float* __restrict__ q, const float* __restrict__ kf,
                         const float* __restrict__ ak, const int* __restrict__ tgt,
                         float* __restrict__ logits, unsigned* __restrict__ segmax,
                         int E) {
    __shared__ float s_ak[HC];
    if (threadIdx.x < HC) s_ak[threadIdx.x] = ak[threadIdx.x];
    __syncthreads();

    const int lane = threadIdx.x & 31;
    const int wave = threadIdx.x >> 5;
    const int hf   = lane >> 4;    // which half of the wave
    const int n    = lane & 15;    // column (head) for B/D, edge-in-tile for A

    // Build all 16 B fragments once. Unconditional LDS load + value select
    // (v_cndmask), NOT a predicated load -> no exec-mask churn in the loop.
    v2f bfrag[16];
    #pragma unroll
    for (int kk = 0; kk < 16; ++kk) {
        const int kg0 = 4 * kk + 2 * hf;     // k_local = j + 2*half
        const int kg1 = kg0 + 1;
        const float b0 = s_ak[(kg0 & 7) * HH + (kg0 >> 3)];
        const float b1 = s_ak[(kg1 & 7) * HH + (kg1 >> 3)];
        v2f b;
        b.x = (n == (kg0 >> 3)) ? b0 : 0.0f;
        b.y = (n == (kg1 >> 3)) ? b1 : 0.0f;
        bfrag[kk] = b;
    }

    const int nTiles = (E + 15) >> 4;
    const int nWaves = gridDim.x * 8;

    for (int tile = blockIdx.x * 8 + wave; tile < nTiles; tile += nWaves) {
        const int e0 = tile * 16;
        if (e0 + 16 <= E) {
            v8f acc = {};
            const int rowBase = (e0 + n) * HC;   // A: lane n loads edge e0+n
            #pragma unroll
            for (int kk = 0; kk < 16; ++kk) {
                const int col = 4 * kk + 2 * hf;
                const v2f qv = *(const v2f*)(q  + rowBase + col);
                const v2f kv = *(const v2f*)(kf + rowBase + col);
                v2f a; a.x = qv.x + kv.x; a.y = qv.y + kv.y;
                acc = __builtin_amdgcn_wmma_f32_16x16x4_f32(
                    false, a, false, bfrag[kk], (short)0, acc, false, false);
            }
            // D layout: VGPR r -> M = r + 8*half, N = lane%16. Heads at n<8.
            if (n < HH) {
                #pragma unroll
                for (int r = 0; r < 8; ++r) {
                    const int e = e0 + r + 8 * hf;
                    const float lg = acc[r];
                    logits[e * HH + n] = lg;
                    atomicMax(&segmax[tgt[e] * HH + n], fmap(lg));
                }
            }
        } else {
            // scalar tail (E % 16 != 0); no WMMA here so divergence is fine
            const int cnt = (E - e0) * HH;
            for (int idx = lane; idx < cnt; idx += 32) {
                const int e = e0 + idx / HH;
                const int h = idx % HH;
                float lg = 0.0f;
                #pragma unroll
                for (int c = 0; c < CC; ++c)
                    lg += (q[e * HC + h * CC + c] + kf[e * HC + h * CC + c]) * s_ak[c * HH + h];
                logits[e * HH + h] = lg;
                atomicMax(&segmax[tgt[e] * HH + h], fmap(lg));
            }
        }
    }
}

// ---------------- K3: ex = exp(logit - segmax), denom += ex --------------
__global__ void k_softmax(const int* __restrict__ tgt, float* __restrict__ lex,
                          const unsigned* __restrict__ segmax,
                          float* __restrict__ denom, int E) {
    const int e = blockIdx.x * blockDim.x + threadIdx.x;
    if (e >= E) return;
    const int t = tgt[e];
    #pragma unroll
    for (int h = 0; h < HH; ++h) {
        const float m  = funmap(segmax[t * HH + h]);
        const float ex = __expf(lex[e * HH + h] - m);
        lex[e * HH + h] = ex;
        atomicAdd(&denom[t * HH + h], ex);
    }
}

// ---------------- K4: weighted scatter of q into output ------------------
__global__ void k_scatter(const float* __restrict__ q, const int* __restrict__ tgt,
                          const float* __restrict__ ex, const float* __restrict__ denom,
                          float* __restrict__ out, int EH) {
    const int id = blockIdx.x * blockDim.x + threadIdx.x;
    if (id >= EH) return;
    const int e = id >> 3;
    const int h = id & 7;
    const int t = tgt[e];
    const float attn = ex[id] / denom[t * HH + h];
    const float* qp = q + e * HC + h * CC;
    float* op = out + t * HC + h * CC;
    #pragma unroll
    for (int c = 0; c < CC; ++c) atomicAdd(&op[c], qp[c] * attn);
}

// ---------------- K5: in-place relu --------------------------------------
__global__ void k_relu(float* __restrict__ out, int n) {
    const int i = blockIdx.x * blockDim.x + threadIdx.x;
    if (i < n) out[i] = fmaxf(out[i], 0.0f);
}

extern "C" void kernel_launch(void* const* d_in, const int* in_sizes, int n_in,
                              void* d_out, int out_size, void* d_ws, size_t ws_size,
                              hipStream_t stream) {
    const float* q  = (const float*)d_in[0];
    const float* kf = (const float*)d_in[1];
    const float* ak = (const float*)d_in[2];
    const int*  tgt = (const int*)d_in[3];
    float* out = (float*)d_out;

    const int E = in_sizes[3];
    const int N = out_size / HC;

    // workspace: [logits/ex: E*H f32][segmax: N*H u32][denom: N*H f32]
    float*    logits = (float*)d_ws;
    unsigned* segmax = (unsigned*)(logits + (size_t)E * HH);
    float*    denom  = (float*)(segmax + (size_t)N * HH);

    const int nNH = N * HH;
    k_init<<<(out_size + 255) / 256, 256, 0, stream>>>(out, segmax, denom, out_size, nNH);

    // 8 waves/block, ~4 tiles per wave via grid-stride
    const int tiles = (E + 15) / 16;
    int blocks = (tiles + 31) / 32;
    if (blocks < 1) blocks = 1;
    k_logits<<<blocks, 256, 0, stream>>>(q, kf, ak, tgt, logits, segmax, E);

    k_softmax<<<(E + 255) / 256, 256, 0, stream>>>(tgt, logits, segmax, denom, E);

    const int EH = E * HH;
    k_scatter<<<(EH + 255) / 256, 256, 0, stream>>>(q, tgt, logits, denom, out, EH);

    k_relu<<<(out_size + 255) / 256, 256, 0, stream>>>(out, out_size);
}